// LPGNN_12970801234329
// MI455X (gfx1250) — compile-verified
//
#include <hip/hip_runtime.h>
#include <hip/hip_bf16.h>

typedef __attribute__((ext_vector_type(2))) float v2f;
typedef __attribute__((ext_vector_type(8))) float v8f;

#define DF 128
#define HID 32

// ---------- init MLP: [nf -> 32 -> 128] on stacked scalar node features ----------
__global__ void init_mlp_kernel(const float* __restrict__ f0,
                                const float* __restrict__ f1,
                                const float* __restrict__ f2,
                                int nf,
                                const float* __restrict__ W0, const float* __restrict__ b0,
                                const float* __restrict__ W1, const float* __restrict__ b1,
                                float* __restrict__ out) {
  int row = blockIdx.x;
  int d = threadIdx.x;
  float x0 = f0[row];
  float x1 = f1[row];
  float x2 = (nf > 2) ? f2[row] : 0.0f;
  float acc = b1[d];
  for (int k = 0; k < HID; ++k) {
    float h = b0[k] + x0 * W0[k] + x1 * W0[HID + k];
    if (nf > 2) h += x2 * W0[2 * HID + k];
    h = fmaxf(h, 0.0f);
    acc += h * W1[k * DF + d];
  }
  out[(size_t)row * DF + d] = acc;
}

// ---------- MLP [128 -> 32 -> 128], one block (128 thr) per row ----------
__global__ void mlp_d_kernel(const float* __restrict__ X,
                             const float* __restrict__ W0, const float* __restrict__ b0,
                             const float* __restrict__ W1, const float* __restrict__ b1,
                             float* __restrict__ Y) {
  __shared__ float xs[DF];
  __shared__ float hp[4 * HID];
  __shared__ float hs[HID];
  int row = blockIdx.x;
  int tid = threadIdx.x;
  xs[tid] = X[(size_t)row * DF + tid];
  __syncthreads();
  int part = tid >> 5, k = tid & 31;
  float ps = 0.0f;
  int j0 = part * 32;
  for (int j = 0; j < 32; ++j) ps += xs[j0 + j] * W0[(j0 + j) * HID + k];
  hp[part * HID + k] = ps;
  __syncthreads();
  if (tid < HID) {
    float h = b0[tid] + hp[tid] + hp[HID + tid] + hp[2 * HID + tid] + hp[3 * HID + tid];
    hs[tid] = fmaxf(h, 0.0f);
  }
  __syncthreads();
  float acc = b1[tid];
  for (int kk = 0; kk < HID; ++kk) acc += hs[kk] * W1[kk * DF + tid];
  Y[(size_t)row * DF + tid] = acc;
}

// ---------- concat MLP [256 -> 32 -> 128], safe for Y == X1 (in-place) ----------
__global__ void mlp_concat_kernel(const float* __restrict__ X1,
                                  const float* __restrict__ X2,
                                  const float* __restrict__ W0, const float* __restrict__ b0,
                                  const float* __restrict__ W1, const float* __restrict__ b1,
                                  float* __restrict__ Y) {
  __shared__ float xs[2 * DF];
  __shared__ float hp[4 * HID];
  __shared__ float hs[HID];
  int row = blockIdx.x;
  int tid = threadIdx.x;
  xs[tid] = X1[(size_t)row * DF + tid];
  xs[DF + tid] = X2[(size_t)row * DF + tid];
  __syncthreads();
  int part = tid >> 5, k = tid & 31;
  float ps = 0.0f;
  int j0 = part * 64;
  for (int j = 0; j < 64; ++j) ps += xs[j0 + j] * W0[(j0 + j) * HID + k];
  hp[part * HID + k] = ps;
  __syncthreads();
  if (tid < HID) {
    float h = b0[tid] + hp[tid] + hp[HID + tid] + hp[2 * HID + tid] + hp[3 * HID + tid];
    hs[tid] = fmaxf(h, 0.0f);
  }
  __syncthreads();
  float acc = b1[tid];
  for (int kk = 0; kk < HID; ++kk) acc += hs[kk] * W1[kk * DF + tid];
  Y[(size_t)row * DF + tid] = acc;
}

// ---------- s_v[b] = E[b] (MxN) @ Bm[b] (NxDF) via fp32 WMMA 16x16x4 ----------
// block 64 (2 waves). Each wave: one 16-row m-tile x 64 cols = 4 accumulators,
// reusing the A fragment across 4 WMMAs. Invalid rows: clamped address +
// per-lane zero select (no EXEC manipulation in the hot loop).
// grid(ceil(M/16), B)
__global__ void gemm_ev_kernel(const float* __restrict__ E,   // [B,M,N]
                               const float* __restrict__ Bm,  // [B,N,DF]
                               float* __restrict__ C,         // [B,M,DF]
                               int M, int N) {
  int lane = threadIdx.x & 31;
  int wave = threadIdx.x >> 5;
  int d0 = wave * 64;              // this wave covers cols d0 .. d0+63
  int m0 = blockIdx.x * 16;
  int bb = blockIdx.y;
  const float* Eb = E + (size_t)bb * M * N;
  const float* Bb = Bm + (size_t)bb * N * DF;
  int mrow = lane & 15;            // A-fragment row (M)
  int koff = (lane >> 4) * 2;      // A/B-fragment K sub-offset
  int ncol = lane & 15;            // B/C-fragment column
  bool mvalid = (m0 + mrow) < M;
  int mclamp = mvalid ? (m0 + mrow) : 0;
  const float* arow = Eb + (size_t)mclamp * N;
  v8f acc0 = {}, acc1 = {}, acc2 = {}, acc3 = {};
  for (int k = 0; k < N; k += 4) {
    v2f a = *(const v2f*)(arow + k + koff);   // 8B-aligned unconditional load
    a.x = mvalid ? a.x : 0.0f;                // per-lane select, no EXEC churn
    a.y = mvalid ? a.y : 0.0f;
    const float* brow0 = Bb + (size_t)(k + koff) * DF + d0 + ncol;
    const float* brow1 = brow0 + DF;
    v2f b0f, b1f, b2f, b3f;
    b0f.x = brow0[0];   b0f.y = brow1[0];
    b1f.x = brow0[16];  b1f.y = brow1[16];
    b2f.x = brow0[32];  b2f.y = brow1[32];
    b3f.x = brow0[48];  b3f.y = brow1[48];
    acc0 = __builtin_amdgcn_wmma_f32_16x16x4_f32(false, a, false, b0f, (short)0, acc0, false, false);
    acc1 = __builtin_amdgcn_wmma_f32_16x16x4_f32(false, a, false, b1f, (short)0, acc1, false, false);
    acc2 = __builtin_amdgcn_wmma_f32_16x16x4_f32(false, a, false, b2f, (short)0, acc2, false, false);
    acc3 = __builtin_amdgcn_wmma_f32_16x16x4_f32(false, a, false, b3f, (short)0, acc3, false, false);
  }
  int mhalf = (lane >> 4) * 8;
  for (int r = 0; r < 8; ++r) {
    int m = m0 + mhalf + r;
    if (m < M) {
      float* crow = C + ((size_t)bb * M + m) * DF + d0 + ncol;
      crow[0]  = acc0[r];
      crow[16] = acc1[r];
      crow[32] = acc2[r];
      crow[48] = acc3[r];
    }
  }
}

// ---------- s_w[b] = E[b]^T (NxM) @ Bm[b] (MxDF) via fp32 WMMA 16x16x4 ----------
// block 64 (2 waves), each wave 16 rows x 64 cols (4 accumulators).
// grid(ceil(N/16), B)
__global__ void gemm_et_kernel(const float* __restrict__ E,   // [B,M,N]
                               const float* __restrict__ Bm,  // [B,M,DF]
                               float* __restrict__ C,         // [B,N,DF]
                               int M, int N) {
  int lane = threadIdx.x & 31;
  int wave = threadIdx.x >> 5;
  int d0 = wave * 64;
  int jj0 = blockIdx.x * 16;
  int bb = blockIdx.y;
  const float* Eb = E + (size_t)bb * M * N;
  const float* Bb = Bm + (size_t)bb * M * DF;
  int jrow = lane & 15;
  int koff = (lane >> 4) * 2;
  int ncol = lane & 15;
  bool jvalid = (jj0 + jrow) < N;
  int jclamp = jvalid ? (jj0 + jrow) : 0;
  v8f acc0 = {}, acc1 = {}, acc2 = {}, acc3 = {};
  for (int k = 0; k < M; k += 4) {
    const float* ecol = Eb + (size_t)(k + koff) * N + jclamp;
    v2f a;
    a.x = ecol[0];
    a.y = ecol[N];
    a.x = jvalid ? a.x : 0.0f;
    a.y = jvalid ? a.y : 0.0f;
    const float* brow0 = Bb + (size_t)(k + koff) * DF + d0 + ncol;
    const float* brow1 = brow0 + DF;
    v2f b0f, b1f, b2f, b3f;
    b0f.x = brow0[0];   b0f.y = brow1[0];
    b1f.x = brow0[16];  b1f.y = brow1[16];
    b2f.x = brow0[32];  b2f.y = brow1[32];
    b3f.x = brow0[48];  b3f.y = brow1[48];
    acc0 = __builtin_amdgcn_wmma_f32_16x16x4_f32(false, a, false, b0f, (short)0, acc0, false, false);
    acc1 = __builtin_amdgcn_wmma_f32_16x16x4_f32(false, a, false, b1f, (short)0, acc1, false, false);
    acc2 = __builtin_amdgcn_wmma_f32_16x16x4_f32(false, a, false, b2f, (short)0, acc2, false, false);
    acc3 = __builtin_amdgcn_wmma_f32_16x16x4_f32(false, a, false, b3f, (short)0, acc3, false, false);
  }
  int jhalf = (lane >> 4) * 8;
  for (int r = 0; r < 8; ++r) {
    int j = jj0 + jhalf + r;
    if (j < N) {
      float* crow = C + ((size_t)bb * N + j) * DF + d0 + ncol;
      crow[0]  = acc0[r];
      crow[16] = acc1[r];
      crow[32] = acc2[r];
      crow[48] = acc3[r];
    }
  }
}

// ---------- pooled[b] = [sum_i hv[b,i,:], sum_j hw[b,j,:]] ----------
__global__ void pool_kernel(const float* __restrict__ hv, const float* __restrict__ hw,
                            float* __restrict__ pooled, int M, int N) {
  int bb = blockIdx.x;
  int t = threadIdx.x;  // 0..255
  float s = 0.0f;
  if (t < DF) {
    const float* p = hv + (size_t)bb * M * DF + t;
    for (int i = 0; i < M; ++i) s += p[(size_t)i * DF];
  } else {
    const float* p = hw + (size_t)bb * N * DF + (t - DF);
    for (int j = 0; j < N; ++j) s += p[(size_t)j * DF];
  }
  pooled[bb * 256 + t] = s;
}

// ---------- final MLP [384 -> 32 -> 1], one wave per output element ----------
__global__ void out_mlp_kernel(const float* __restrict__ pooled, // [B,256]
                               const float* __restrict__ hw,     // [B,N,128]
                               const float* __restrict__ W0,     // [384,32]
                               const float* __restrict__ b0,     // [32]
                               const float* __restrict__ W1,     // [32,1]
                               const float* __restrict__ b1,     // [1]
                               float* __restrict__ out, int N, int totalRows) {
  int wave = threadIdx.x >> 5;
  int lane = threadIdx.x & 31;
  int row = blockIdx.x * 4 + wave;
  if (row >= totalRows) return;
  int bb = row / N;
  const float* pl = pooled + bb * 256;
  const float* xr = hw + (size_t)row * DF;
  float h = b0[lane];
  for (int t = 0; t < 256; ++t) h += pl[t] * W0[t * HID + lane];
  for (int d = 0; d < DF; ++d) h += xr[d] * W0[(256 + d) * HID + lane];
  h = fmaxf(h, 0.0f);
  float p = h * W1[lane];
  for (int off = 16; off > 0; off >>= 1) p += __shfl_down(p, off, 32);
  if (lane == 0) out[row] = p + b1[0];
}

extern "C" void kernel_launch(void* const* d_in, const int* in_sizes, int n_in,
                              void* d_out, int out_size, void* d_ws, size_t ws_size,
                              hipStream_t stream) {
  (void)in_sizes; (void)n_in; (void)out_size; (void)ws_size;
  const int Bc = 128, Mc = 200, Nc = 1000;
  const float* c_in = (const float*)d_in[0];
  const float* A    = (const float*)d_in[1];
  const float* bvec = (const float*)d_in[2];
  const float* cons = (const float*)d_in[3];
  const float* lvec = (const float*)d_in[4];
  const float* uvec = (const float*)d_in[5];
  auto P = [&](int i) { return (const float*)d_in[i]; };
  // params flattened in setup_inputs insertion order, each MLP layer as (W, b):
  // fv_in: 6..9   fw_in: 10..13   fv[l]: 14+4l   gv[l]: 30+4l
  // fw[l]: 46+4l  gw[l]: 62+4l    f_out: 78..81 (unused)   fw_out: 82..85

  float* ws     = (float*)d_ws;
  float* hv     = ws;                                   // [B,M,DF]
  float* hw     = hv + (size_t)Bc * Mc * DF;            // [B,N,DF]
  float* tbig   = hw + (size_t)Bc * Nc * DF;            // [B,N,DF] (t_w / s_w)
  float* s_v    = tbig + (size_t)Bc * Nc * DF;          // [B,M,DF]
  float* t_v    = s_v + (size_t)Bc * Mc * DF;           // [B,M,DF]
  float* pooled = t_v + (size_t)Bc * Mc * DF;           // [B,256]

  // init_features
  init_mlp_kernel<<<Bc * Mc, DF, 0, stream>>>(bvec, cons, nullptr, 2,
      P(6), P(7), P(8), P(9), hv);
  init_mlp_kernel<<<Bc * Nc, DF, 0, stream>>>(c_in, lvec, uvec, 3,
      P(10), P(11), P(12), P(13), hw);

  for (int l = 0; l < 4; ++l) {
    int fv = 14 + 4 * l, gv = 30 + 4 * l, fw = 46 + 4 * l, gw = 62 + 4 * l;
    // t_w = fw(hw)
    mlp_d_kernel<<<Bc * Nc, DF, 0, stream>>>(hw, P(fw), P(fw + 1), P(fw + 2), P(fw + 3), tbig);
    // s_v = E @ t_w
    gemm_ev_kernel<<<dim3((Mc + 15) / 16, Bc), 64, 0, stream>>>(A, tbig, s_v, Mc, Nc);
    // t_v = fv(old hv)  (must precede the in-place hv update)
    mlp_d_kernel<<<Bc * Mc, DF, 0, stream>>>(hv, P(fv), P(fv + 1), P(fv + 2), P(fv + 3), t_v);
    // hv = gv([hv, s_v])   (in-place safe: row staged in LDS)
    mlp_concat_kernel<<<Bc * Mc, DF, 0, stream>>>(hv, s_v, P(gv), P(gv + 1), P(gv + 2), P(gv + 3), hv);
    // s_w = E^T @ t_v   (reuse tbig)
    gemm_et_kernel<<<dim3((Nc + 15) / 16, Bc), 64, 0, stream>>>(A, t_v, tbig, Mc, Nc);
    // hw = gw([hw, s_w])  (in-place safe)
    mlp_concat_kernel<<<Bc * Nc, DF, 0, stream>>>(hw, tbig, P(gw), P(gw + 1), P(gw + 2), P(gw + 3), hw);
  }

  pool_kernel<<<Bc, 256, 0, stream>>>(hv, hw, pooled, Mc, Nc);
  out_mlp_kernel<<<(Bc * Nc + 3) / 4, 128, 0, stream>>>(pooled, hw,
      P(82), P(83), P(84), P(85), (float*)d_out, Nc, Bc * Nc);
}